// TokenMixer_29068338659807
// MI455X (gfx1250) — compile-verified
//
#include <hip/hip_runtime.h>

typedef __attribute__((ext_vector_type(2))) float v2f;
typedef __attribute__((ext_vector_type(8))) float v8f;

#define DIM    256
#define RED    64
#define KKERN  9
#define HH     64
#define WW     64
#define HW     4096          // HH*WW
#define BATCH  8
#define TILE_H 4

// ---------------------------------------------------------------------------
// Depthwise 3x3 conv, zero 'SAME' padding. One block = (batch, channel, 4-row
// tile); stages a (4+2)x(64+2) halo tile of the channel plane in LDS.
// ---------------------------------------------------------------------------
__global__ __launch_bounds__(256) void dwconv3x3_kernel(
    const float* __restrict__ in, const float* __restrict__ wgt,
    const float* __restrict__ bias, float* __restrict__ out) {
  const int blk = blockIdx.x;
  const int rt = blk % (HH / TILE_H);
  const int c  = (blk / (HH / TILE_H)) % DIM;
  const int b  = blk / ((HH / TILE_H) * DIM);
  const int h0 = rt * TILE_H;

  __shared__ float tile[TILE_H + 2][WW + 2];
  const float* plane = in + (size_t)(b * DIM + c) * HW;
  const int tid = threadIdx.x;

  for (int idx = tid; idx < (TILE_H + 2) * (WW + 2); idx += 256) {
    const int r   = idx / (WW + 2);
    const int col = idx % (WW + 2);
    const int gh = h0 + r - 1;
    const int gw = col - 1;
    float v = 0.0f;
    if (gh >= 0 && gh < HH && gw >= 0 && gw < WW) v = plane[gh * WW + gw];
    tile[r][col] = v;
  }
  __syncthreads();

  const int tx = tid & 63;   // 0..63 = full row
  const int ty = tid >> 6;   // 0..3

  float wv[9];
#pragma unroll
  for (int t = 0; t < 9; ++t) wv[t] = wgt[c * 9 + t];

  float acc = bias[c];
#pragma unroll
  for (int i = 0; i < 3; ++i)
#pragma unroll
    for (int j = 0; j < 3; ++j)
      acc = fmaf(wv[i * 3 + j], tile[ty + i][tx + j], acc);

  out[(size_t)(b * DIM + c) * HW + (h0 + ty) * WW + tx] = acc;
}

// ---------------------------------------------------------------------------
// Fused: kern = w_span @ relu(w_red @ x1 + b_red) + b_span, per pixel.
// One wave computes the full 64-row reduced vector for 16 pixels with
// V_WMMA_F32_16X16X4_F32 (4 accumulator tiles, 64 K-steps), then collapses
// to the 9 span outputs in-register (cross-half shfl_xor combine).
//
// f32 16x16x4 fragment layouts (ISA 7.12.2):
//   A (16x4):  lane%16 = M,  K = 2*(lane/16) + vgpr
//   B (4x16):  lane%16 = N,  K = 2*(lane/16) + vgpr
//   C/D:       lane%16 = N,  M = vgpr + 8*(lane/16)
// ---------------------------------------------------------------------------
__global__ __launch_bounds__(256) void reduce_span_kernel(
    const float* __restrict__ x1, const float* __restrict__ w_red,
    const float* __restrict__ b_red, const float* __restrict__ w_span,
    const float* __restrict__ b_span, float* __restrict__ kern) {
  const int wave = (blockIdx.x * 256 + threadIdx.x) >> 5;   // 0..2047
  const int lane = threadIdx.x & 31;
  const int hi   = lane >> 4;     // K-pair / M-half selector
  const int l16  = lane & 15;

  const int b  = wave >> 8;                 // 256 pixel-tiles per batch
  const int p0 = (wave & 255) * 16;         // first pixel of this tile
  const float* xb = x1 + (size_t)b * DIM * HW;

  v8f acc0 = {}, acc1 = {}, acc2 = {}, acc3 = {};

  for (int k0 = 0; k0 < DIM; k0 += 4) {
    const int ka = k0 + 2 * hi;            // even -> 8B-aligned v2f loads of A

    // B fragment: two channel slices of x1 for this pixel (stride HW apart)
    v2f bf;
    bf[0] = xb[(size_t)ka * HW + p0 + l16];
    bf[1] = xb[(size_t)(ka + 1) * HW + p0 + l16];

    // prefetch next K-slice (speculative; dropped if OOB)
    __builtin_prefetch(&xb[(size_t)(ka + 4) * HW + p0 + l16], 0, 1);

    // A fragments: rows of w_red for the 4 output tiles (consecutive k pair)
    const v2f a0 = *(const v2f*)(w_red + (size_t)( 0 + l16) * DIM + ka);
    const v2f a1 = *(const v2f*)(w_red + (size_t)(16 + l16) * DIM + ka);
    const v2f a2 = *(const v2f*)(w_red + (size_t)(32 + l16) * DIM + ka);
    const v2f a3 = *(const v2f*)(w_red + (size_t)(48 + l16) * DIM + ka);

    acc0 = __builtin_amdgcn_wmma_f32_16x16x4_f32(false, a0, false, bf,
                                                 (short)0, acc0, false, false);
    acc1 = __builtin_amdgcn_wmma_f32_16x16x4_f32(false, a1, false, bf,
                                                 (short)0, acc1, false, false);
    acc2 = __builtin_amdgcn_wmma_f32_16x16x4_f32(false, a2, false, bf,
                                                 (short)0, acc2, false, false);
    acc3 = __builtin_amdgcn_wmma_f32_16x16x4_f32(false, a3, false, bf,
                                                 (short)0, acc3, false, false);
  }

  // Unpack: this lane holds rows {t*16 + 8*hi + r} of y for pixel p0+l16.
  float yv[32];
#pragma unroll
  for (int r = 0; r < 8; ++r) {
    yv[r]      = acc0[r];
    yv[8 + r]  = acc1[r];
    yv[16 + r] = acc2[r];
    yv[24 + r] = acc3[r];
  }

  float partial[KKERN];
#pragma unroll
  for (int j = 0; j < KKERN; ++j) partial[j] = 0.0f;

#pragma unroll
  for (int t = 0; t < 4; ++t) {
#pragma unroll
    for (int r = 0; r < 8; ++r) {
      const int row = t * 16 + 8 * hi + r;
      float y = fmaxf(yv[t * 8 + r] + b_red[row], 0.0f);   // bias + ReLU
#pragma unroll
      for (int j = 0; j < KKERN; ++j)
        partial[j] = fmaf(w_span[j * RED + row], y, partial[j]);
    }
  }

  // Each pixel's 64 rows are split between lane p and lane p+16: combine.
#pragma unroll
  for (int j = 0; j < KKERN; ++j) {
    const float full = partial[j] + __shfl_xor(partial[j], 16, 32);
    if (lane < 16)
      kern[(size_t)(b * KKERN + j) * HW + p0 + l16] = full + b_span[j];
  }
}

// ---------------------------------------------------------------------------
// Involution apply: out[b,c,h,w] = sum_k kern[b,k,h,w] * x1[b,c,h+di,w+dj].
// Same LDS plane-tiling as the depthwise conv; the 9 dynamic weights are
// shared across all 256 channels -> pure L2 hits after the first channel.
// ---------------------------------------------------------------------------
__global__ __launch_bounds__(256) void invol_apply_kernel(
    const float* __restrict__ x1, const float* __restrict__ kern,
    float* __restrict__ out) {
  const int blk = blockIdx.x;
  const int rt = blk % (HH / TILE_H);
  const int c  = (blk / (HH / TILE_H)) % DIM;
  const int b  = blk / ((HH / TILE_H) * DIM);
  const int h0 = rt * TILE_H;

  __shared__ float tile[TILE_H + 2][WW + 2];
  const float* plane = x1 + (size_t)(b * DIM + c) * HW;
  const int tid = threadIdx.x;

  for (int idx = tid; idx < (TILE_H + 2) * (WW + 2); idx += 256) {
    const int r   = idx / (WW + 2);
    const int col = idx % (WW + 2);
    const int gh = h0 + r - 1;
    const int gw = col - 1;
    float v = 0.0f;
    if (gh >= 0 && gh < HH && gw >= 0 && gw < WW) v = plane[gh * WW + gw];
    tile[r][col] = v;
  }
  __syncthreads();

  const int tx = tid & 63;
  const int ty = tid >> 6;
  const int p  = (h0 + ty) * WW + tx;

  float acc = 0.0f;
#pragma unroll
  for (int i = 0; i < 3; ++i)
#pragma unroll
    for (int j = 0; j < 3; ++j) {
      const float kw = kern[(size_t)(b * KKERN + (i * 3 + j)) * HW + p];
      acc = fmaf(kw, tile[ty + i][tx + j], acc);
    }

  out[(size_t)(b * DIM + c) * HW + p] = acc;
}

// ---------------------------------------------------------------------------
extern "C" void kernel_launch(void* const* d_in, const int* in_sizes, int n_in,
                              void* d_out, int out_size, void* d_ws,
                              size_t ws_size, hipStream_t stream) {
  const float* x      = (const float*)d_in[0];
  const float* w_in   = (const float*)d_in[1];
  const float* b_in   = (const float*)d_in[2];
  const float* w_red  = (const float*)d_in[3];
  const float* b_red  = (const float*)d_in[4];
  const float* w_span = (const float*)d_in[5];
  const float* b_span = (const float*)d_in[6];
  const float* w_out  = (const float*)d_in[7];
  const float* b_out  = (const float*)d_in[8];
  float* out = (float*)d_out;

  const size_t X1_BYTES   = (size_t)BATCH * DIM * HW * sizeof(float);   // 33.55 MB
  const size_t KERN_BYTES = (size_t)BATCH * KKERN * HW * sizeof(float); //  1.18 MB
  char* ws = (char*)d_ws;
  float* x1   = (float*)ws;
  float* kern = (float*)(ws + X1_BYTES);
  float* x2   = (float*)(ws + X1_BYTES + KERN_BYTES);

  const dim3 blk(256);
  const int nconv = BATCH * DIM * (HH / TILE_H);   // 32768 blocks
  const int ngemm = (BATCH * HW / 16) / 8;         // 2048 waves / 8 per block

  dwconv3x3_kernel<<<nconv, blk, 0, stream>>>(x, w_in, b_in, x1);
  reduce_span_kernel<<<ngemm, blk, 0, stream>>>(x1, w_red, b_red,
                                                w_span, b_span, kern);
  invol_apply_kernel<<<nconv, blk, 0, stream>>>(x1, kern, x2);
  dwconv3x3_kernel<<<nconv, blk, 0, stream>>>(x2, w_out, b_out, out);
}